// EAS_GCN_41154376630515
// MI455X (gfx1250) — compile-verified
//
#include <hip/hip_runtime.h>
#include <hip/hip_bf16.h>
#include <math.h>

typedef __attribute__((ext_vector_type(2))) float v2f;
typedef __attribute__((ext_vector_type(8))) float v8f;

#define TPB 256

// ---------------------------------------------------------------------------
// Elementwise helpers
// ---------------------------------------------------------------------------
__global__ void k_fill(float* __restrict__ p, float v, int n) {
    int i = blockIdx.x * blockDim.x + threadIdx.x;
    if (i < n) p[i] = v;
}

__global__ void k_deg_edges(const int* __restrict__ dst, float* __restrict__ deg, int E) {
    int e = blockIdx.x * blockDim.x + threadIdx.x;
    if (e < E) unsafeAtomicAdd(&deg[dst[e]], 1.0f);
}

__global__ void k_rsqrt(const float* __restrict__ deg, float* __restrict__ dinv, int n) {
    int i = blockIdx.x * blockDim.x + threadIdx.x;
    if (i < n) dinv[i] = rsqrtf(deg[i]);
}

// ---------------------------------------------------------------------------
// Pack W[K,M] (row-major) into WMMA B-fragment order, zero-padded to Mp cols:
//   Wp[((k/2)*Mp + c)*2 + {0,1}] = W[k*M + c], W[(k+1)*M + c]   (0 if c >= M)
// so a lane fetches its (k+2h, k+2h+1) pair with ONE coalesced b64 load.
// ---------------------------------------------------------------------------
__global__ void k_pack_w(const float* __restrict__ W, float* __restrict__ Wp,
                         int K, int M, int Mp)
{
    int i = blockIdx.x * blockDim.x + threadIdx.x;
    int total = (K >> 1) * Mp;
    if (i >= total) return;
    int kp = i / Mp;
    int c  = i - kp * Mp;
    float v0 = 0.0f, v1 = 0.0f;
    if (c < M) {
        v0 = W[(size_t)(2 * kp) * M + c];
        v1 = W[(size_t)(2 * kp + 1) * M + c];
    }
    Wp[(size_t)i * 2 + 0] = v0;
    Wp[(size_t)i * 2 + 1] = v1;
}

// ---------------------------------------------------------------------------
// WMMA fp32 GEMM:  C[N,M] = act(A[N,K] @ W[K,M] + bias), W pre-packed (Wp, Mp).
// Requires N % 16 == 0, K % 4 == 0, Mp % 64 == 0 (cols zero-padded in Wp).
// Each wave: 16 rows x 64 cols (4 tiles of 16x16).  8 waves / block.
// Branch-free inner loop: one b64 A load + four b64 B loads + four WMMAs.
// ---------------------------------------------------------------------------
__global__ __launch_bounds__(TPB)
void k_gemm_wmma(const float* __restrict__ A, const float* __restrict__ Wp,
                 const float* __restrict__ bias, float* __restrict__ C,
                 int N, int K, int M, int Mp, int relu)
{
    const int wave = threadIdx.x >> 5;
    const int lane = threadIdx.x & 31;
    const int row0 = (blockIdx.x * 8 + wave) << 4;
    if (row0 >= N) return;
    const int col0 = blockIdx.y << 6;
    const int half = lane >> 4;     // 0: lanes 0-15, 1: lanes 16-31
    const int l15  = lane & 15;

    v8f acc[4];
#pragma unroll
    for (int t = 0; t < 4; ++t) acc[t] = v8f{};

    // A-frag: lane holds row (l15), K-pair 2*half .. 2*half+1 (b64 load)
    const float* arow = A + (size_t)(row0 + l15) * K + (half << 1);
    // B-frag: packed pairs, lane's column = col0 + t*16 + l15
    const v2f* Bp = (const v2f*)Wp;
    const int colb = col0 + l15;

    for (int k = 0; k < K; k += 4) {
        v2f a = *(const v2f*)(arow + k);
        const v2f* brow = Bp + (size_t)((k >> 1) + half) * Mp + colb;
#pragma unroll
        for (int t = 0; t < 4; ++t) {
            v2f b = brow[t << 4];
            acc[t] = __builtin_amdgcn_wmma_f32_16x16x4_f32(
                /*neg_a=*/false, a, /*neg_b=*/false, b,
                /*c_mod=*/(short)0, acc[t],
                /*reuse_a=*/false, /*reuse_b=*/false);
        }
    }

    // Epilogue: bias + optional relu.  C/D layout: VGPR v -> row v + 8*half.
#pragma unroll
    for (int t = 0; t < 4; ++t) {
        const int col = col0 + (t << 4) + l15;
        if (col >= M) continue;
        const float bv = bias[col];
#pragma unroll
        for (int v = 0; v < 8; ++v) {
            const int row = row0 + v + (half << 3);
            float val = acc[t][v] + bv;
            if (relu) val = fmaxf(val, 0.0f);
            C[(size_t)row * M + col] = val;
        }
    }
}

// GEMV for M == 1 (gnd layer): out[n] = dot(A[n,:], W) + b0
__global__ void k_gemv1(const float* __restrict__ A, const float* __restrict__ W,
                        const float* __restrict__ b, float* __restrict__ out,
                        int N, int K)
{
    int n = blockIdx.x * blockDim.x + threadIdx.x;
    if (n >= N) return;
    const float* a = A + (size_t)n * K;
    float s = 0.0f;
    for (int k = 0; k < K; ++k) s = fmaf(a[k], W[k], s);
    out[n] = s + b[0];
}

// ---------------------------------------------------------------------------
// Edge aggregation: agg[dst] += xw[src] * dinv[src]*dinv[dst]
// One thread per (edge, feature).
// ---------------------------------------------------------------------------
__global__ void k_edge_agg(const float* __restrict__ xw, const int* __restrict__ src,
                           const int* __restrict__ dst, const float* __restrict__ dinv,
                           float* __restrict__ agg, int E, int F)
{
    int tid = blockIdx.x * blockDim.x + threadIdx.x;
    int total = E * F;
    if (tid >= total) return;
    int e = tid / F;
    int f = tid - e * F;
    int s = src[e], d = dst[e];
    float w = dinv[s] * dinv[d];
    unsafeAtomicAdd(&agg[(size_t)d * F + f], xw[(size_t)s * F + f] * w);
}

// h = agg + xw * (1/deg);  mix variant:  out = (1-p)*h + p*tra1
__global__ void k_finish_mix(const float* __restrict__ agg, const float* __restrict__ xw,
                             const float* __restrict__ tra1, const float* __restrict__ dinv,
                             float* __restrict__ out, int N, int F, float p)
{
    int tid = blockIdx.x * blockDim.x + threadIdx.x;
    if (tid >= N * F) return;
    int n = tid / F;
    float id = dinv[n] * dinv[n];   // 1/deg
    float h = agg[tid] + xw[tid] * id;
    out[tid] = (1.0f - p) * h + p * tra1[tid];
}

__global__ void k_finish_plain(const float* __restrict__ agg, const float* __restrict__ xw,
                               const float* __restrict__ dinv, float* __restrict__ out,
                               int N, int F)
{
    int tid = blockIdx.x * blockDim.x + threadIdx.x;
    if (tid >= N * F) return;
    int n = tid / F;
    float id = dinv[n] * dinv[n];
    out[tid] = agg[tid] + xw[tid] * id;
}

// ---------------------------------------------------------------------------
// log-softmax over 32 classes: one wave (32 lanes) per row.
// ---------------------------------------------------------------------------
__global__ void k_log_softmax32(const float* __restrict__ h, float* __restrict__ out, int N)
{
    int gtid = blockIdx.x * blockDim.x + threadIdx.x;
    int row  = gtid >> 5;
    int lane = gtid & 31;
    if (row >= N) return;
    float v = h[(size_t)row * 32 + lane];
    float mx = v;
#pragma unroll
    for (int off = 16; off > 0; off >>= 1) mx = fmaxf(mx, __shfl_xor(mx, off, 32));
    float ex = expf(v - mx);
    float sm = ex;
#pragma unroll
    for (int off = 16; off > 0; off >>= 1) sm += __shfl_xor(sm, off, 32);
    out[(size_t)row * 32 + lane] = (v - mx) - logf(sm);
}

// ---------------------------------------------------------------------------
// Orchestration
// ---------------------------------------------------------------------------
static inline int padM(int M) { return ((M + 63) / 64) * 64; }
static inline dim3 gemm_grid(int N, int M) {
    return dim3((unsigned)((N + 127) / 128), (unsigned)((M + 63) / 64), 1);
}

extern "C" void kernel_launch(void* const* d_in, const int* in_sizes, int n_in,
                              void* d_out, int out_size, void* d_ws, size_t ws_size,
                              hipStream_t stream)
{
    const float* x      = (const float*)d_in[0];
    const int*   src    = (const int*)  d_in[1];
    const int*   dst    = (const int*)  d_in[2];
    const float* enc1_W = (const float*)d_in[3];
    const float* enc1_b = (const float*)d_in[4];
    const float* cls_W  = (const float*)d_in[5];
    const float* cls_b  = (const float*)d_in[6];
    const float* dec1_W = (const float*)d_in[7];
    const float* dec1_b = (const float*)d_in[8];
    const float* xbar_W = (const float*)d_in[9];
    const float* xbar_b = (const float*)d_in[10];
    const float* lin1_W = (const float*)d_in[11];
    const float* lin1_b = (const float*)d_in[12];
    const float* lin2_W = (const float*)d_in[13];
    const float* lin2_b = (const float*)d_in[14];
    const float* g1m_W  = (const float*)d_in[15];
    const float* g1m_b  = (const float*)d_in[16];
    const float* g2_W   = (const float*)d_in[17];
    const float* g2_b   = (const float*)d_in[18];
    const float* gnd_W  = (const float*)d_in[19];
    const float* gnd_b  = (const float*)d_in[20];

    const int N = in_sizes[0] / 500;   // n_input = 500
    const int E = in_sizes[1];
    const float P_MIX = 0.5f;
    const size_t Ns = (size_t)N;

    float* out = (float*)d_out;
    float* xbar_out = out;                 // [N,500]  (written last)
    float* pred_out = out + 500 * Ns;      // [N,32]
    float* nd_out   = out + 532 * Ns;      // [N,1]

    // d_ws layout (floats): deg(N) dinv(N) tra1(96N) z(64N) buf1(96N) | packed W
    float* ws   = (float*)d_ws;
    float* deg  = ws;
    float* dinv = ws + Ns;
    float* tra1 = ws + 2 * Ns;
    float* z    = tra1 + 96 * Ns;
    float* buf1 = z + 64 * Ns;             // xw1 / hmix / dec_h1

    // packed weights (K * padM(M) floats each), ~1.2 MB total
    float* pk = buf1 + 96 * Ns;
    float* p_enc1 = pk;                    pk += 500 * padM(96);
    float* p_cls  = pk;                    pk += 96  * padM(64);
    float* p_dec1 = pk;                    pk += 64  * padM(96);
    float* p_xbar = pk;                    pk += 96  * padM(500);
    float* p_lin1 = pk;                    pk += 500 * padM(256);
    float* p_lin2 = pk;                    pk += 256 * padM(128);
    float* p_g1m  = pk;                    pk += 128 * padM(96);
    float* p_g2   = pk;                    pk += 96  * padM(32);

    // x_bar region of d_out used as scratch until the final GEMM:
    float* m1   = out;                     // [N,256]
    float* m2   = out + 256 * Ns;          // [N,128]
    float* agg1 = out + 384 * Ns;          // [N,96]
    float* xw2  = out;                     // [N,32]  (m1 dead by then)
    float* agg2 = out + 32 * Ns;           // [N,32]
    float* xwz  = out + 64 * Ns;           // [N,1]
    float* aggz = out + 65 * Ns;           // [N,1]

    auto blocks = [](int n) { return (n + TPB - 1) / TPB; };
    auto pack = [&](const float* W, float* Wp, int K, int M) {
        int total = (K >> 1) * padM(M);
        k_pack_w<<<blocks(total), TPB, 0, stream>>>(W, Wp, K, M, padM(M));
    };

    // --- pack all weights (tiny, once per launch) ---
    pack(enc1_W, p_enc1, 500, 96);
    pack(cls_W,  p_cls,  96,  64);
    pack(dec1_W, p_dec1, 64,  96);
    pack(xbar_W, p_xbar, 96,  500);
    pack(lin1_W, p_lin1, 500, 256);
    pack(lin2_W, p_lin2, 256, 128);
    pack(g1m_W,  p_g1m,  128, 96);
    pack(g2_W,   p_g2,   96,  32);

    // --- degrees ---
    k_fill<<<blocks(N), TPB, 0, stream>>>(deg, 1.0f, N);          // self-loop
    k_deg_edges<<<blocks(E), TPB, 0, stream>>>(dst, deg, E);
    k_rsqrt<<<blocks(N), TPB, 0, stream>>>(deg, dinv, N);

    // --- AE front: tra1, z ---
    k_gemm_wmma<<<gemm_grid(N, 96), TPB, 0, stream>>>(x, p_enc1, enc1_b, tra1, N, 500, 96, padM(96), 1);
    k_gemm_wmma<<<gemm_grid(N, 64), TPB, 0, stream>>>(tra1, p_cls, cls_b, z, N, 96, 64, padM(64), 0);

    // --- MLP branch ---
    k_gemm_wmma<<<gemm_grid(N, 256), TPB, 0, stream>>>(x, p_lin1, lin1_b, m1, N, 500, 256, padM(256), 1);
    k_gemm_wmma<<<gemm_grid(N, 128), TPB, 0, stream>>>(m1, p_lin2, lin2_b, m2, N, 256, 128, padM(128), 0);

    // --- GCN layer 1: xw1 = m2 @ g1m + b ---
    k_gemm_wmma<<<gemm_grid(N, 96), TPB, 0, stream>>>(m2, p_g1m, g1m_b, buf1, N, 128, 96, padM(96), 0);
    k_fill<<<blocks(N * 96), TPB, 0, stream>>>(agg1, 0.0f, N * 96);
    k_edge_agg<<<blocks(E * 96), TPB, 0, stream>>>(buf1, src, dst, dinv, agg1, E, 96);
    k_finish_mix<<<blocks(N * 96), TPB, 0, stream>>>(agg1, buf1, tra1, dinv, buf1, N, 96, P_MIX);

    // --- GCN layer 2: xw2 = hmix @ g2 + b ---
    k_gemm_wmma<<<gemm_grid(N, 32), TPB, 0, stream>>>(buf1, p_g2, g2_b, xw2, N, 96, 32, padM(32), 0);
    k_fill<<<blocks(N * 32), TPB, 0, stream>>>(agg2, 0.0f, N * 32);
    k_edge_agg<<<blocks(E * 32), TPB, 0, stream>>>(xw2, src, dst, dinv, agg2, E, 32);
    k_finish_plain<<<blocks(N * 32), TPB, 0, stream>>>(agg2, xw2, dinv, xw2, N, 32);
    k_log_softmax32<<<blocks(N * 32), TPB, 0, stream>>>(xw2, pred_out, N);

    // --- GCN nd head (F = 1) ---
    k_gemv1<<<blocks(N), TPB, 0, stream>>>(z, gnd_W, gnd_b, xwz, N, 64);
    k_fill<<<blocks(N), TPB, 0, stream>>>(aggz, 0.0f, N);
    k_edge_agg<<<blocks(E), TPB, 0, stream>>>(xwz, src, dst, dinv, aggz, E, 1);
    k_finish_plain<<<blocks(N), TPB, 0, stream>>>(aggz, xwz, dinv, nd_out, N, 1);

    // --- AE tail (writes x_bar last, freeing the scratch region) ---
    k_gemm_wmma<<<gemm_grid(N, 96), TPB, 0, stream>>>(z, p_dec1, dec1_b, buf1, N, 64, 96, padM(96), 1);
    k_gemm_wmma<<<gemm_grid(N, 500), TPB, 0, stream>>>(buf1, p_xbar, xbar_b, xbar_out, N, 96, 500, padM(500), 0);
}